// CRF_40278203301966
// MI455X (gfx1250) — compile-verified
//
#include <hip/hip_runtime.h>
#include <hip/hip_bf16.h>

typedef __attribute__((ext_vector_type(2))) float v2f;
typedef __attribute__((ext_vector_type(8))) float v8f;

namespace {
constexpr int kB = 512;
constexpr int kS = 1024;
constexpr int kT = 48;
constexpr int kNT = 3;   // 48 tags = 3 N-tiles of 16
constexpr int kKT = 12;  // 48 tags = 12 K-tiles of 4
}

// Forward (log-partition) scan. One wave (32 lanes) owns 16 batch rows.
// Scores live in WMMA C-layout (3 x v8f). Each step:
//   rmax -> P=exp(score-rmax) -> LDS bounce to A-layout -> 36x WMMA f32 16x16x4
//   -> score = log(acc) + rmax + emit(t) (pipelined in registers), masked select.
// Single wave per block => no barriers needed: LDS ops from one wave are
// processed in order, so the store->load bounce is safe with only scheduling
// fences. emissions/mask for step t+1 are loaded during step t; step t+2 is
// prefetched into the WGP cache.
__global__ __launch_bounds__(32) void crf_forward_48t(
    const float* __restrict__ emissions,
    const float* __restrict__ transitions,
    const float* __restrict__ start_t,
    const float* __restrict__ end_t,
    const unsigned char* __restrict__ mask,
    float* __restrict__ lp_out)
{
  const int lane = threadIdx.x;
  const int half = lane >> 4;   // 0: lanes 0-15, 1: lanes 16-31
  const int col  = lane & 15;
  const int b0   = blockIdx.x * 16;

  __shared__ float Pbuf[16 * kT];

  // exp(transitions) resident in registers, B-fragment (4x16 f32) layout:
  // VGPR0: lanes0-15 K=kb+0, lanes16-31 K=kb+2; VGPR1: K=kb+1 / K=kb+3.
  v2f Bf[kNT][kKT];
#pragma unroll
  for (int nt = 0; nt < kNT; ++nt) {
#pragma unroll
    for (int kt = 0; kt < kKT; ++kt) {
      const int n  = nt * 16 + col;
      const int k0 = kt * 4 + 2 * half;
      Bf[nt][kt].x = __expf(transitions[(k0 + 0) * kT + n]);
      Bf[nt][kt].y = __expf(transitions[(k0 + 1) * kT + n]);
    }
  }

  // Per-row base offsets (row m = r + 8*half of this 16-row tile).
  size_t erow[8], mrow[8];
#pragma unroll
  for (int r = 0; r < 8; ++r) {
    const int m = r + 8 * half;
    erow[r] = (size_t)(b0 + m) * (kS * kT);
    mrow[r] = (size_t)(b0 + m) * kS;
  }

  // score0 = start + emissions[:,0]  (C layout: elem r -> row m=r+8*half, tag j=nt*16+col)
  v8f score[kNT];
#pragma unroll
  for (int nt = 0; nt < kNT; ++nt) {
#pragma unroll
    for (int r = 0; r < 8; ++r) {
      score[nt][r] = start_t[nt * 16 + col] + emissions[erow[r] + nt * 16 + col];
    }
  }

  // Software pipeline: registers hold emissions/mask for the step being consumed.
  float eReg[kNT][8];
  unsigned char mReg[8];
#pragma unroll
  for (int r = 0; r < 8; ++r) {
    mReg[r] = mask[mrow[r] + 1];
#pragma unroll
    for (int nt = 0; nt < kNT; ++nt)
      eReg[nt][r] = emissions[erow[r] + (size_t)kT + nt * 16 + col];
  }

  for (int t = 1; t < kS; ++t) {
    // per-row max over all 48 tags (cross-lane max within 16-lane group)
    float rmax[8];
#pragma unroll
    for (int r = 0; r < 8; ++r) {
      float mx = fmaxf(fmaxf(score[0][r], score[1][r]), score[2][r]);
#pragma unroll
      for (int off = 8; off >= 1; off >>= 1)
        mx = fmaxf(mx, __shfl_xor(mx, off, 32));
      rmax[r] = mx;
    }

    // P = exp(score - rmax) -> LDS row-major [m][j]
#pragma unroll
    for (int nt = 0; nt < kNT; ++nt) {
#pragma unroll
      for (int r = 0; r < 8; ++r) {
        Pbuf[(r + 8 * half) * kT + nt * 16 + col] = __expf(score[nt][r] - rmax[r]);
      }
    }
    // Single wave: LDS is processed in order; only block instruction motion.
    __builtin_amdgcn_sched_barrier(0);

    // A fragments (16x4 f32): lane row = col, K pair = kt*4 + 2*half
    v2f Af[kKT];
#pragma unroll
    for (int kt = 0; kt < kKT; ++kt) {
      const float* src = &Pbuf[col * kT + kt * 4 + 2 * half];
      Af[kt].x = src[0];
      Af[kt].y = src[1];
    }
    // Keep all 6 ds_load_2addr_b64 grouped -> one DS wait before the WMMAs.
    __builtin_amdgcn_sched_barrier(0);

    // acc[nt] = P x exp(T) over 12 K-tiles (3 independent accumulator chains)
    v8f acc[kNT] = {};
#pragma unroll
    for (int nt = 0; nt < kNT; ++nt) {
#pragma unroll
      for (int kt = 0; kt < kKT; ++kt) {
        acc[nt] = __builtin_amdgcn_wmma_f32_16x16x4_f32(
            false, Af[kt], false, Bf[nt][kt], (short)0, acc[nt], false, false);
      }
    }

    // score = log(acc) + rmax + emissions[:, t] (from pipeline regs), masked
#pragma unroll
    for (int r = 0; r < 8; ++r) {
      const unsigned char mv = mReg[r];
#pragma unroll
      for (int nt = 0; nt < kNT; ++nt) {
        const float nxt = __logf(acc[nt][r]) + rmax[r] + eReg[nt][r];
        score[nt][r] = mv ? nxt : score[nt][r];
      }
    }

    // Refill pipeline for step t+1; WGP-scope prefetch for step t+2.
    const int tn = (t + 1 < kS) ? (t + 1) : (kS - 1);
#pragma unroll
    for (int r = 0; r < 8; ++r) {
      mReg[r] = mask[mrow[r] + tn];
      __builtin_prefetch(&emissions[erow[r] + (size_t)tn * kT + kT + col], 0, 3);
#pragma unroll
      for (int nt = 0; nt < kNT; ++nt)
        eReg[nt][r] = emissions[erow[r] + (size_t)tn * kT + nt * 16 + col];
    }
  }

  // log_partition = logsumexp_j(score + end)
  float fin[kNT][8];
#pragma unroll
  for (int nt = 0; nt < kNT; ++nt)
#pragma unroll
    for (int r = 0; r < 8; ++r)
      fin[nt][r] = score[nt][r] + end_t[nt * 16 + col];
#pragma unroll
  for (int r = 0; r < 8; ++r) {
    float mx = fmaxf(fmaxf(fin[0][r], fin[1][r]), fin[2][r]);
#pragma unroll
    for (int off = 8; off >= 1; off >>= 1)
      mx = fmaxf(mx, __shfl_xor(mx, off, 32));
    float s = 0.f;
#pragma unroll
    for (int nt = 0; nt < kNT; ++nt) s += __expf(fin[nt][r] - mx);
#pragma unroll
    for (int off = 8; off >= 1; off >>= 1)
      s += __shfl_xor(s, off, 32);
    if (col == 0) lp_out[b0 + r + 8 * half] = mx + __logf(s);
  }
}

// Gold-path score: one thread per batch row.
__global__ void crf_gold_48t(
    const float* __restrict__ emissions,
    const float* __restrict__ transitions,
    const float* __restrict__ start_t,
    const float* __restrict__ end_t,
    const int* __restrict__ tags,
    const unsigned char* __restrict__ mask,
    float* __restrict__ gold_out)
{
  const int b = blockIdx.x * blockDim.x + threadIdx.x;
  if (b >= kB) return;
  const size_t tb = (size_t)b * kS;
  const size_t eb = (size_t)b * kS * kT;
  int prev = tags[tb];
  float g = start_t[prev] + emissions[eb + prev];
  int cnt = mask[tb] ? 1 : 0;
  for (int t = 1; t < kS; ++t) {
    const int cur = tags[tb + t];
    if (mask[tb + t]) {
      ++cnt;
      g += transitions[prev * kT + cur] + emissions[eb + (size_t)t * kT + cur];
    }
    prev = cur;
  }
  g += end_t[tags[tb + (cnt - 1)]];
  gold_out[b] = g;
}

// mean(log_partition - gold) over 512 rows.
__global__ void crf_reduce_mean(const float* __restrict__ lp,
                                const float* __restrict__ gold,
                                float* __restrict__ out)
{
  __shared__ float sdata[256];
  const int tid = threadIdx.x;
  float v = (lp[tid] - gold[tid]) + (lp[tid + 256] - gold[tid + 256]);
  sdata[tid] = v;
  __syncthreads();
#pragma unroll
  for (int s = 128; s > 0; s >>= 1) {
    if (tid < s) sdata[tid] += sdata[tid + s];
    __syncthreads();
  }
  if (tid == 0) out[0] = sdata[0] / (float)kB;
}

extern "C" void kernel_launch(void* const* d_in, const int* in_sizes, int n_in,
                              void* d_out, int out_size, void* d_ws, size_t ws_size,
                              hipStream_t stream) {
  const float* emissions          = (const float*)d_in[0];
  const float* transitions        = (const float*)d_in[1];
  const float* start_transitions  = (const float*)d_in[2];
  const float* end_transitions    = (const float*)d_in[3];
  const int*   tags               = (const int*)d_in[4];
  const unsigned char* mask       = (const unsigned char*)d_in[5];

  float* lp   = (float*)d_ws;       // 512 floats
  float* gold = lp + kB;            // 512 floats

  crf_forward_48t<<<kB / 16, 32, 0, stream>>>(
      emissions, transitions, start_transitions, end_transitions, mask, lp);
  crf_gold_48t<<<kB / 256, 256, 0, stream>>>(
      emissions, transitions, start_transitions, end_transitions, tags, mask, gold);
  crf_reduce_mean<<<1, 256, 0, stream>>>(lp, gold, (float*)d_out);
}